// MultiTimeAttention_18829136625682
// MI455X (gfx1250) — compile-verified
//
#include <hip/hip_runtime.h>
#include <math.h>
#include <stdint.h>

// ---------------------------------------------------------------------------
// MultiTimeAttention on gfx1250 (MI455X), fp32 end-to-end via
// V_WMMA_F32_16X16X4_F32.  Shapes: b=8, q=k=256, emb=128, h=4, ek=32, f=32,
// hidden=64.  The per-feature masked softmax is restructured (shift
// invariance) into:  e = exp(S - rowmax);  num = e @ (mask*V);
// den = e @ mask;  out = num/den  -- turning the whole pipeline into GEMMs.
// Problem is ~0.5 GFLOP / ~10 MB -> L2-resident + latency bound on MI455X,
// so fp32 WMMA (full reference precision) is the right precision choice.
// ---------------------------------------------------------------------------

typedef float v2f __attribute__((ext_vector_type(2)));
typedef float v8f __attribute__((ext_vector_type(8)));

// D = A(16x4, f32) * B(4x16, f32) + C(16x16, f32)
// A frag: lane l holds A[l%16][2*(l/16)+v], v in {0,1}
// B frag: lane l holds B[2*(l/16)+v][l%16]
// C/D   : lane l, reg r holds C[r + 8*(l/16)][l%16]
__device__ __forceinline__ v8f wmma_k4(v2f a, v2f b, v8f c) {
  return __builtin_amdgcn_wmma_f32_16x16x4_f32(
      /*neg_a=*/false, a, /*neg_b=*/false, b,
      /*c_mod=*/(short)0, c, /*reuse_a=*/false, /*reuse_b=*/false);
}

// ASYNCcnt-tracked copy of 16 contiguous bytes (per lane) global -> LDS.
// ldsoff = wave-relative LDS byte offset (low 32 bits of generic pointer).
__device__ __forceinline__ void async_g2lds_b128(uint32_t ldsoff, uint64_t gaddr) {
  asm volatile("global_load_async_to_lds_b128 %0, %1, off"
               :: "v"(ldsoff), "v"(gaddr) : "memory");
}
__device__ __forceinline__ void wait_async0() {
  asm volatile("s_wait_asynccnt 0x0" ::: "memory");
}

// ---------------------------------------------------------------------------
// C[M,N] = A[M,K] * W[N,K]^T (+ bias).   grid = (1, M/16); block covers one
// 16-row A tile shared by (N/16) waves.  A tile (16xK, K<=128) is staged
// into LDS once per block via async-to-LDS (8x reuse for the projections).
// ---------------------------------------------------------------------------
__global__ __launch_bounds__(256)
void gemm16_wmma(const float* __restrict__ A, const float* __restrict__ W,
                 const float* __restrict__ bias, float* __restrict__ C,
                 int N, int K) {
  __shared__ float Atile[16 * 128];            // 8 KB, K <= 128

  const int lane = threadIdx.x & 31;
  const int wave = threadIdx.x >> 5;
  const int mt = blockIdx.y;                   // M tile
  const int nt = blockIdx.x * (blockDim.x >> 5) + wave;   // N tile (per wave)

  // ---- stage A tile: 16*K contiguous floats -> LDS, b128 per lane ----
  {
    const float* src = A + (size_t)mt * 16 * K;
    const int nvec = (16 * K) >> 2;            // 16B transfers
    for (int t = threadIdx.x; t < nvec; t += blockDim.x) {
      const uint32_t ldsoff = (uint32_t)(uintptr_t)(&Atile[t << 2]);
      async_g2lds_b128(ldsoff, (uint64_t)(uintptr_t)(src + (t << 2)));
    }
  }
  wait_async0();                               // my async writes landed
  __syncthreads();                             // everyone's writes visible

  const int lm = lane & 15;                    // M index (A) / N index (B)
  const int kh = (lane >> 4) << 1;             // K sub-offset: 0 or 2
  const float* arow = &Atile[lm * K];
  const float* wrow = W + (size_t)(nt * 16 + lm) * K;

  v8f acc = {};
  #pragma unroll 8
  for (int k0 = 0; k0 < K; k0 += 4) {
    v2f a, b;
    a.x = arow[k0 + kh];
    a.y = arow[k0 + kh + 1];
    b.x = wrow[k0 + kh];
    b.y = wrow[k0 + kh + 1];
    acc = wmma_k4(a, b, acc);
  }

  const int col = nt * 16 + lm;
  const float bb = bias ? bias[col] : 0.0f;
  const int rbase = (lane >> 4) << 3;          // row offset 0 or 8
  float* cptr = C + (size_t)(mt * 16 + rbase) * N + col;
  #pragma unroll
  for (int r = 0; r < 8; ++r)
    cptr[(size_t)r * N] = acc[r] + bb;
}

// ---------------------------------------------------------------------------
// Fused attention: one wave per (b, h, 16-row q tile).
//   S(16x256) = Q_tile @ K_h^T * scale        (WMMA; row-max tracked in regs)
//   rowmax: 4-step shfl_xor butterfly across the 16-lane half
//   e = exp(S - rowmax)                       (2M exps total, not 67M)
//   num = e @ (mask ? V : 0),  den = e @ mask (WMMA)
//   O[b, q, h*32+f] = num/den
// ---------------------------------------------------------------------------
__global__ __launch_bounds__(32)
void attn_wmma(const float* __restrict__ Qb, const float* __restrict__ Kb,
               const float* __restrict__ V, const int* __restrict__ Msk,
               float* __restrict__ O) {
  __shared__ float S[16 * 256];    // 16 KB score/exp tile
  __shared__ float rmax[16];

  const int lane = threadIdx.x;
  const int lm = lane & 15;
  const int hi = lane >> 4;        // 0 or 1
  const int kh = hi << 1;          // K sub-offset 0 or 2

  const int qt = blockIdx.x & 15;          // q tile (16 rows)
  const int bh = blockIdx.x >> 4;
  const int h  = bh & 3;
  const int b  = bh >> 2;
  const float scale = 0.17677669529663687f;   // 1/sqrt(32)

  const float* Qrow  = Qb + (size_t)(b * 256 + qt * 16 + lm) * 128 + h * 32;
  const float* Kbase = Kb + (size_t)(b * 256) * 128 + h * 32;

  // ---- phase 1: scores + per-lane partial row max ----
  float pmax[8];
  #pragma unroll
  for (int r = 0; r < 8; ++r) pmax[r] = -3.4e38f;

  for (int kt = 0; kt < 16; ++kt) {
    const float* Krow = Kbase + (size_t)(kt * 16 + lm) * 128;
    v8f acc = {};
    #pragma unroll
    for (int e0 = 0; e0 < 32; e0 += 4) {
      v2f a, bf;
      a.x  = Qrow[e0 + kh];  a.y  = Qrow[e0 + kh + 1];
      bf.x = Krow[e0 + kh];  bf.y = Krow[e0 + kh + 1];   // B[e][kcol]=K[kcol][e]
      acc = wmma_k4(a, bf, acc);
    }
    #pragma unroll
    for (int r = 0; r < 8; ++r) {
      const float s = acc[r] * scale;
      pmax[r] = fmaxf(pmax[r], s);
      S[(r + hi * 8) * 256 + kt * 16 + lm] = s;
    }
  }

  // ---- phase 2: butterfly row max (halves stay disjoint for masks<16) ----
  #pragma unroll
  for (int r = 0; r < 8; ++r) {
    float m = pmax[r];
    m = fmaxf(m, __shfl_xor(m, 1, 32));
    m = fmaxf(m, __shfl_xor(m, 2, 32));
    m = fmaxf(m, __shfl_xor(m, 4, 32));
    m = fmaxf(m, __shfl_xor(m, 8, 32));
    pmax[r] = m;                     // row (r + 8*hi) max, all 16 lanes agree
  }
  if (lm == 0) {
    #pragma unroll
    for (int r = 0; r < 8; ++r) rmax[r + hi * 8] = pmax[r];
  }
  __syncthreads();
  for (int i = lane; i < 16 * 256; i += 32)
    S[i] = __expf(S[i] - rmax[i >> 8]);
  __syncthreads();

  // ---- phase 3: num = e @ V', den = e @ mask, out = num/den ----
  const float* Vb = V   + (size_t)b * 256 * 32;
  const int*   Mb = Msk + (size_t)b * 256 * 32;
  float* Orow = O + (size_t)(b * 256 + qt * 16) * 128 + h * 32;

  #pragma unroll
  for (int ft = 0; ft < 2; ++ft) {           // f tiles: [0..15], [16..31]
    const int f = ft * 16 + lm;
    v8f accN = {}, accD = {};
    for (int k0 = 0; k0 < 256; k0 += 4) {
      v2f a, bn, bd;
      a.x = S[lm * 256 + k0 + kh];
      a.y = S[lm * 256 + k0 + kh + 1];
      const int kd0 = k0 + kh, kd1 = kd0 + 1;
      const int m0 = Mb[kd0 * 32 + f];
      const int m1 = Mb[kd1 * 32 + f];
      bn.x = m0 ? Vb[kd0 * 32 + f] : 0.0f;
      bn.y = m1 ? Vb[kd1 * 32 + f] : 0.0f;
      bd.x = m0 ? 1.0f : 0.0f;
      bd.y = m1 ? 1.0f : 0.0f;
      accN = wmma_k4(a, bn, accN);
      accD = wmma_k4(a, bd, accD);
    }
    #pragma unroll
    for (int r = 0; r < 8; ++r)
      Orow[(size_t)(r + hi * 8) * 128 + ft * 16 + lm] = accN[r] / accD[r];
  }
}

// ---------------------------------------------------------------------------
// Host-side launcher
// inputs: 0 query(8,256,128) 1 key(8,256,128) 2 value(8,256,32)
//         3 mask(8,256,32)   4 Wq(128,128)    5 bq(128)
//         6 Wk(128,128)      7 Wv(64,128)     8 bv(64)
// output: (8,256,64) fp32
// ---------------------------------------------------------------------------
extern "C" void kernel_launch(void* const* d_in, const int* in_sizes, int n_in,
                              void* d_out, int out_size, void* d_ws, size_t ws_size,
                              hipStream_t stream) {
  const float* query = (const float*)d_in[0];
  const float* key   = (const float*)d_in[1];
  const float* value = (const float*)d_in[2];
  const int*   mask  = (const int*)  d_in[3];
  const float* Wq    = (const float*)d_in[4];
  const float* bq    = (const float*)d_in[5];
  const float* Wk    = (const float*)d_in[6];
  const float* Wv    = (const float*)d_in[7];
  const float* bv    = (const float*)d_in[8];
  float* out = (float*)d_out;

  float* Qb = (float*)d_ws;            // 2048 x 128
  float* Kb = Qb + 2048 * 128;         // 2048 x 128
  float* Ob = Kb + 2048 * 128;         // 2048 x 128 (heads concatenated)

  // Q = query @ Wq^T + bq : block = 8 waves (N/16), grid = (1, 128)
  gemm16_wmma<<<dim3(1, 128), 256, 0, stream>>>(query, Wq, bq, Qb, 128, 128);
  // K = key @ Wk^T
  gemm16_wmma<<<dim3(1, 128), 256, 0, stream>>>(key, Wk, nullptr, Kb, 128, 128);
  // fused masked attention: 8 b * 4 h * 16 q-tiles = 512 waves
  attn_wmma<<<512, 32, 0, stream>>>(Qb, Kb, value, mask, Ob);
  // out = O @ Wv^T + bv : block = 4 waves (N/16 = 4), grid = (1, 128)
  gemm16_wmma<<<dim3(1, 128), 128, 0, stream>>>(Ob, Wv, bv, out, 64, 128);
}